// CCL__69277822485245
// MI455X (gfx1250) — compile-verified
//
#include <hip/hip_runtime.h>
#include <math.h>

// ---------------------------------------------------------------------------
// DCT-conv for MI455X (gfx1250): all stages as f32 WMMA GEMMs + async
// global->LDS DMA (GLOBAL_LOAD_ASYNC_TO_LDS_B128, ASYNCcnt-tracked).
//   x      : (8, 64, 128, 128)  f32
//   weight : (128, 64, 5, 5)    f32
//   bias   : (128,)             f32
//   out    : (8, 128, 128, 128) f32
//
// Pipeline (exact linear-algebra rewrite of the reference):
//   1) build D (fwd DCT-II) and M (inv DCT) 128x128 matrices
//   2) wd[k,i,j,o] = sum_m D[k,m] * weight[o,i,m,j]   (j padded 5->8 w/ zeros)
//   3) xd = D @ x   along H                           (WMMA f32 16x16x4)
//   4) hd[b,o,k,w] = sum_{i,j} wd[k,i,j,o]*xd[b,i,k,(w-j)&127] -> d_out
//      (implicit GEMM; async-DMA double buffering overlapped with WMMA)
//   5) d_out = M @ d_out + bias  (in-place per 128x128 block via 64KB LDS)
//
// Roofline: ~20 GFLOP f32, ~370 MB HBM traffic (16 us @ 23.3 TB/s) ->
// compute-limited on the f32 matrix pipe, hence V_WMMA_F32_16X16X4_F32.
// Workspace (floats): D[16384] | M[16384] | xd[8*64*128*128] | wd[128*64*8*128]
//   total ~67.2 MB.
// ---------------------------------------------------------------------------

typedef __attribute__((ext_vector_type(2))) float v2f;
typedef __attribute__((ext_vector_type(8))) float v8f;

__device__ __forceinline__ v8f wmma4(v2f a, v2f b, v8f c) {
  // V_WMMA_F32_16X16X4_F32 : D(16x16,f32) = A(16x4,f32) x B(4x16,f32) + C
  return __builtin_amdgcn_wmma_f32_16x16x4_f32(
      /*neg_a=*/false, a, /*neg_b=*/false, b,
      /*c_mod=*/(short)0, c, /*reuse_a=*/false, /*reuse_b=*/false);
}

// Async DMA: global memory -> LDS, 16B per lane, no VGPR round-trip.
// VDST carries the LDS byte address (= low 32 bits of the generic pointer,
// per the ISA aperture rule).
__device__ __forceinline__ void async_load_b128(void* lds_ptr, const void* gptr) {
  unsigned lds_off = (unsigned)(unsigned long long)lds_ptr;
  asm volatile("global_load_async_to_lds_b128 %0, %1, off"
               :: "v"(lds_off), "v"(gptr) : "memory");
}
__device__ __forceinline__ void wait_async0() {
  asm volatile("s_wait_asynccnt 0x0" ::: "memory");
}

#define ZERO8 {0.f,0.f,0.f,0.f,0.f,0.f,0.f,0.f}

// ---------------------------------------------------------------------------
// Stage 1: DCT matrices. D[k,n] = 2*cos(pi*(2n+1)k/256),
// M[n,k] = cos(pi*(2n+1)k/256) * (k==0 ? 0.5 : 1) / 256.
// ---------------------------------------------------------------------------
__global__ void build_dct_kernel(float* __restrict__ Dm, float* __restrict__ Mi) {
  int t = blockIdx.x * 256 + threadIdx.x;   // 0..16383
  int r = t >> 7;
  int c = t & 127;
  double ang1 = M_PI * (2.0 * c + 1.0) * (double)r / 256.0;  // D[r=k][c=n]
  Dm[t] = (float)(2.0 * cos(ang1));
  double ang2 = M_PI * (2.0 * r + 1.0) * (double)c / 256.0;  // M[r=n][c=k]
  Mi[t] = (float)(cos(ang2) * (c == 0 ? 0.5 : 1.0) / 256.0);
}

// ---------------------------------------------------------------------------
// Stage 2: per-H-frequency W-taps. wd[((k*64+i)*8+j)*128+o]
// ---------------------------------------------------------------------------
__global__ void build_wd_kernel(const float* __restrict__ Dm,
                                const float* __restrict__ w,
                                float* __restrict__ wd) {
  size_t t = (size_t)blockIdx.x * 256 + threadIdx.x;  // 8,388,608 total
  int o = (int)(t & 127);
  int j = (int)((t >> 7) & 7);
  int i = (int)((t >> 10) & 63);
  int k = (int)(t >> 16);
  float v = 0.f;
  if (j < 5) {
#pragma unroll
    for (int m = 0; m < 5; ++m)
      v += Dm[k * 128 + m] * w[((o * 64 + i) * 5 + m) * 5 + j];
  }
  wd[t] = v;
}

// ---------------------------------------------------------------------------
// Stage 3: xd[b,c,k,w] = sum_h D[k,h] * x[b,c,h,w].  One workgroup per (b,c).
// 8 waves; wave covers k-rows [16*wave,16*wave+16), all 128 w columns.
// ---------------------------------------------------------------------------
__global__ __launch_bounds__(256) void dct_fwd_kernel(const float* __restrict__ Dm,
                                                      const float* __restrict__ x,
                                                      float* __restrict__ xd) {
  __shared__ __align__(16) float ldsX[128 * 128];   // 64 KB input block
  const int bc = blockIdx.x;                        // b*64 + c
  const float* src = x + (size_t)bc * 16384;
  float* dst = xd + (size_t)bc * 16384;
  for (int idx = threadIdx.x; idx < 4096; idx += 256)
    async_load_b128(&((float4*)ldsX)[idx], &((const float4*)src)[idx]);
  wait_async0();
  __syncthreads();

  const int lane  = threadIdx.x & 31;
  const int wave  = threadIdx.x >> 5;
  const int m15   = lane & 15;
  const int dlt   = (lane >> 4) << 1;   // K-half: lanes16-31 hold K=2,3
  const int kbase = wave * 16;

  v8f acc[8] = {ZERO8, ZERO8, ZERO8, ZERO8, ZERO8, ZERO8, ZERO8, ZERO8};

  for (int h0 = 0; h0 < 128; h0 += 4) {
    v2f a;                              // A[M=k][K=h], M on lanes, K on vgpr+half
    const float* drow = Dm + (kbase + m15) * 128 + h0 + dlt;
    a.x = drow[0];
    a.y = drow[1];
#pragma unroll
    for (int nt = 0; nt < 8; ++nt) {
      v2f b;                            // B[K=h][N=w], N on lanes
      const float* brow = ldsX + (h0 + dlt) * 128 + nt * 16 + m15;
      b.x = brow[0];
      b.y = brow[128];
      acc[nt] = wmma4(a, b, acc[nt]);
    }
  }

  const int rhi = (lane >> 4) << 3;     // C rows: lanes16-31 -> M = r+8
#pragma unroll
  for (int nt = 0; nt < 8; ++nt)
#pragma unroll
    for (int r = 0; r < 8; ++r)
      dst[(kbase + r + rhi) * 128 + nt * 16 + m15] = acc[nt][r];
}

// ---------------------------------------------------------------------------
// Stage 4: per (b,k): C[o, w] = sum_{i,j} wd[k,i,j,o] * xd[b,i,k,(w-j)&127].
// M=o (128), N=w (128), K=i*8+j (512, zero-padded taps).
// Async-DMA double buffering: fire next slice's loads, run 16 WMMAs on the
// current slice, then s_wait_asynccnt 0 + barrier.
// Result hd[b,o,k,w] written to d_out (consumed in-place by stage 5).
// ---------------------------------------------------------------------------
__global__ __launch_bounds__(256) void midconv_kernel(const float* __restrict__ xd,
                                                      const float* __restrict__ wd,
                                                      float* __restrict__ out) {
  __shared__ __align__(16) float xrow[2][128];      // xd[b,i,k,:] for current i
  __shared__ __align__(16) float wsl[2][8 * 128];   // wd[k,i,:,:]
  const int b = blockIdx.x;   // 0..7
  const int k = blockIdx.y;   // 0..127

  const int lane  = threadIdx.x & 31;
  const int wave  = threadIdx.x >> 5;
  const int m15   = lane & 15;
  const int dlt   = (lane >> 4) << 1;
  const int obase = wave * 16;

  v8f acc[8] = {ZERO8, ZERO8, ZERO8, ZERO8, ZERO8, ZERO8, ZERO8, ZERO8};

  auto load_slice = [&](int i, int buf) {
    // xrow: 512 B = 32 lanes x 16 B  (wave 0 only)
    if (threadIdx.x < 32)
      async_load_b128(&xrow[buf][threadIdx.x * 4],
                      xd + (((size_t)b * 64 + i) * 128 + k) * 128 + threadIdx.x * 4);
    // wsl: 4 KB = 256 threads x 16 B
    async_load_b128(&wsl[buf][threadIdx.x * 4],
                    wd + ((size_t)k * 64 + i) * 1024 + threadIdx.x * 4);
  };

  load_slice(0, 0);
  wait_async0();
  __syncthreads();

  for (int i = 0; i < 64; ++i) {
    const int cur = i & 1;
    if (i + 1 < 64) load_slice(i + 1, cur ^ 1);   // fire-and-forget DMA
#pragma unroll
    for (int j0 = 0; j0 < 8; j0 += 4) {
      v2f a;                            // A[M=o][K=j], from wd slice
      a.x = wsl[cur][(j0 + dlt) * 128 + obase + m15];
      a.y = wsl[cur][(j0 + dlt + 1) * 128 + obase + m15];
#pragma unroll
      for (int nt = 0; nt < 8; ++nt) {  // B[K=j][N=w] = shifted xd row
        const int w0 = nt * 16 + m15;
        v2f bb;
        bb.x = xrow[cur][(w0 - (j0 + dlt)) & 127];
        bb.y = xrow[cur][(w0 - (j0 + dlt + 1)) & 127];
        acc[nt] = wmma4(a, bb, acc[nt]);
      }
    }
    wait_async0();     // this wave's DMA for i+1 has landed
    __syncthreads();   // all waves' DMA landed; all reads of old buffer done
  }

  const int rhi = (lane >> 4) << 3;
#pragma unroll
  for (int nt = 0; nt < 8; ++nt)
#pragma unroll
    for (int r = 0; r < 8; ++r)
      out[(((size_t)b * 128 + obase + r + rhi) * 128 + k) * 128 + nt * 16 + m15] =
          acc[nt][r];
}

// ---------------------------------------------------------------------------
// Stage 5: per (b,o) block, in place: out = M @ hd + bias[o].
// hd block staged through 64KB LDS so global writes can't race global reads.
// ---------------------------------------------------------------------------
__global__ __launch_bounds__(256) void idct_bias_kernel(const float* __restrict__ Mi,
                                                        const float* __restrict__ bias,
                                                        float* __restrict__ out) {
  __shared__ __align__(16) float ldsH[128 * 128];
  const int bo = blockIdx.x;            // b*128 + o
  const int o  = bo & 127;
  float* blk = out + (size_t)bo * 16384;
  for (int idx = threadIdx.x; idx < 4096; idx += 256)
    async_load_b128(&((float4*)ldsH)[idx], &((const float4*)blk)[idx]);
  wait_async0();
  __syncthreads();

  const int lane  = threadIdx.x & 31;
  const int wave  = threadIdx.x >> 5;
  const int m15   = lane & 15;
  const int dlt   = (lane >> 4) << 1;
  const int nbase = wave * 16;

  v8f acc[8] = {ZERO8, ZERO8, ZERO8, ZERO8, ZERO8, ZERO8, ZERO8, ZERO8};

  for (int k0 = 0; k0 < 128; k0 += 4) {
    v2f a;                              // A[M=n][K=k] from inverse-DCT matrix
    const float* mrow = Mi + (nbase + m15) * 128 + k0 + dlt;
    a.x = mrow[0];
    a.y = mrow[1];
#pragma unroll
    for (int nt = 0; nt < 8; ++nt) {    // B[K=k][N=w] from LDS block
      v2f b;
      const float* brow = ldsH + (k0 + dlt) * 128 + nt * 16 + m15;
      b.x = brow[0];
      b.y = brow[128];
      acc[nt] = wmma4(a, b, acc[nt]);
    }
  }

  const float bv = bias[o];
  const int rhi = (lane >> 4) << 3;
#pragma unroll
  for (int nt = 0; nt < 8; ++nt)
#pragma unroll
    for (int r = 0; r < 8; ++r)
      blk[(nbase + r + rhi) * 128 + nt * 16 + m15] = acc[nt][r] + bv;
}

// ---------------------------------------------------------------------------
extern "C" void kernel_launch(void* const* d_in, const int* in_sizes, int n_in,
                              void* d_out, int out_size, void* d_ws, size_t ws_size,
                              hipStream_t stream) {
  (void)in_sizes; (void)n_in; (void)out_size; (void)ws_size;
  const float* x      = (const float*)d_in[0];  // (8,64,128,128)
  const float* weight = (const float*)d_in[1];  // (128,64,5,5)
  const float* bias   = (const float*)d_in[2];  // (128,)
  float* out = (float*)d_out;                   // (8,128,128,128)

  float* ws = (float*)d_ws;
  float* Dm = ws;                                   // 16384 floats
  float* Mi = ws + 16384;                           // 16384 floats
  float* xd = ws + 32768;                           // 8*64*128*128
  float* wd = xd + (size_t)8 * 64 * 128 * 128;      // 128*64*8*128

  build_dct_kernel<<<64, 256, 0, stream>>>(Dm, Mi);
  build_wd_kernel<<<32768, 256, 0, stream>>>(Dm, weight, wd);
  dct_fwd_kernel<<<512, 256, 0, stream>>>(Dm, x, xd);
  midconv_kernel<<<dim3(8, 128), 256, 0, stream>>>(xd, wd, out);
  idct_bias_kernel<<<1024, 256, 0, stream>>>(Mi, bias, out);
}